// AdaptiveSampler_58274116272326
// MI455X (gfx1250) — compile-verified
//
#include <hip/hip_runtime.h>
#include <hip/hip_bf16.h>
#include <stdint.h>

typedef __attribute__((ext_vector_type(2))) float v2f;
typedef __attribute__((ext_vector_type(4))) float f4;
typedef __attribute__((ext_vector_type(8))) float v8f;

#define B_  8
#define H_  512
#define W_  512
#define HW_ (H_ * W_)

// ---------------- Kernel 1a: partial per-batch min/max (64 partials/batch) ----
__global__ void __launch_bounds__(256)
k_minmax_part(const float* __restrict__ x, float* __restrict__ pmin, float* __restrict__ pmax) {
    const int b = blockIdx.y, blk = blockIdx.x, t = threadIdx.x;
    const float* p = x + (size_t)b * HW_ + (size_t)blk * 4096;
    float mn = 3.4e38f, mx = -3.4e38f;
#pragma unroll
    for (int i = 0; i < 16; ++i) {
        float v = p[t + i * 256];
        mn = fminf(mn, v); mx = fmaxf(mx, v);
    }
    __shared__ float smn[256], smx[256];
    smn[t] = mn; smx[t] = mx;
    __syncthreads();
    for (int s = 128; s > 0; s >>= 1) {
        if (t < s) { smn[t] = fminf(smn[t], smn[t + s]); smx[t] = fmaxf(smx[t], smx[t + s]); }
        __syncthreads();
    }
    if (t == 0) { pmin[b * 64 + blk] = smn[0]; pmax[b * 64 + blk] = smx[0]; }
}

// ---------------- Kernel 1b: reduce 64 partials -> ws[1024+b], ws[1032+b] -----
__global__ void __launch_bounds__(64)
k_minmax_final(float* __restrict__ ws) {
    const int b = blockIdx.x, t = threadIdx.x;
    __shared__ float smn[64], smx[64];
    smn[t] = ws[b * 64 + t];
    smx[t] = ws[512 + b * 64 + t];
    __syncthreads();
    for (int s = 32; s > 0; s >>= 1) {
        if (t < s) { smn[t] = fminf(smn[t], smn[t + s]); smx[t] = fmaxf(smx[t], smx[t + s]); }
        __syncthreads();
    }
    if (t == 0) { ws[1024 + b] = smn[0]; ws[1032 + b] = smx[0]; }
}

// Banded Gaussian weight: W_band[m,k] = w5[k-m] if 0<=k-m<5 else 0 (select chain,
// stays in VGPRs — no dynamic indexing).
__device__ __forceinline__ float bandv(int d, const float w5[5]) {
    float r = 0.f;
    r = (d == 0) ? w5[0] : r;
    r = (d == 1) ? w5[1] : r;
    r = (d == 2) ? w5[2] : r;
    r = (d == 3) ? w5[3] : r;
    r = (d == 4) ? w5[4] : r;
    return r;
}

// ---------------- Kernel 2: normalize + separable 5x5 Gaussian + density ------
// One wave (32 lanes) per 16x16 output tile. Halo tile 20x20 staged into LDS via
// async global->LDS copies; vertical pass runs on the WMMA pipe:
//   Out(16x16) = W_band(16x20) x H(20x16)  as 5 chained v_wmma_f32_16x16x4_f32.
__global__ void __launch_bounds__(32)
k_density(const float* __restrict__ x, const float* __restrict__ k2,
          const float* __restrict__ ws, float* __restrict__ dens) {
    const int lane = threadIdx.x;
    const int tx = blockIdx.x, ty = blockIdx.y, b = blockIdx.z;
    const int r0 = ty * 16, c0 = tx * 16;

    const float mn = ws[1024 + b];
    const float mx = ws[1032 + b];
    const float inv = (mx > mn) ? 1.0f / (mx - mn) : 0.0f;

    // 1D Gaussian from row-sums of the (separable, sum==1) 5x5 kernel.
    float w5[5];
#pragma unroll
    for (int r = 0; r < 5; ++r) {
        float s = 0.f;
#pragma unroll
        for (int c = 0; c < 5; ++c) s += k2[r * 5 + c];
        w5[r] = s;
    }

    __shared__ float T[20][20];   // normalized halo tile (zero outside image)
    __shared__ float Hh[20][16];  // horizontal-pass result

    const float* src = x + (size_t)b * HW_;

    // Zero only out-of-bounds halo entries (disjoint from async-load targets).
    for (int idx = lane; idx < 400; idx += 32) {
        const int rr = idx / 20, cc = idx % 20;
        const int gr = r0 - 2 + rr, gc = c0 - 2 + cc;
        if ((unsigned)gr >= H_ || (unsigned)gc >= W_) T[rr][cc] = 0.f;
    }
    asm volatile("s_wait_dscnt 0" ::: "memory");

    // Async global->LDS copy of the in-bounds 20x20 region (row-wise, lanes 0..19).
#pragma unroll 1
    for (int rr = 0; rr < 20; ++rr) {
        const int gr = r0 - 2 + rr, gc = c0 - 2 + lane;
        if (lane < 20 && (unsigned)gr < H_ && (unsigned)gc < W_) {
            unsigned lds_off = (unsigned)(uintptr_t)&T[rr][lane];
            const float* g = src + gr * W_ + gc;
            asm volatile("global_load_async_to_lds_b32 %0, %1, off"
                         :: "v"(lds_off), "v"(g) : "memory");
        }
    }
    asm volatile("s_wait_asynccnt 0" ::: "memory");
    __syncthreads();

    // Normalize in place (in-bounds only; OOB entries stay exactly zero).
    for (int idx = lane; idx < 400; idx += 32) {
        const int rr = idx / 20, cc = idx % 20;
        const int gr = r0 - 2 + rr, gc = c0 - 2 + cc;
        if ((unsigned)gr < H_ && (unsigned)gc < W_) T[rr][cc] = (T[rr][cc] - mn) * inv;
    }
    __syncthreads();

    // Horizontal 1D conv on VALU: Hh[r][c] = sum_k w5[k] * T[r][c+k].
    for (int idx = lane; idx < 320; idx += 32) {
        const int r = idx >> 4, c = idx & 15;
        float a = 0.f;
#pragma unroll
        for (int k = 0; k < 5; ++k) a = fmaf(w5[k], T[r][c + k], a);
        Hh[r][c] = a;
    }
    __syncthreads();

    // Vertical 1D conv on the matrix pipe. A(16x4) layout: lanes 0-15 hold M=lane,
    // K=4j,4j+1 in v0,v1; lanes 16-31 hold K=4j+2,4j+3. B(4x16) mirrors it.
    const int m  = lane & 15;      // A row == B/D column for this lane
    const int hi = lane >> 4;
    v8f acc = {};
#pragma unroll
    for (int j = 0; j < 5; ++j) {
        const int kk = 4 * j + 2 * hi;
        v2f a, bb;
        a.x  = bandv(kk     - m, w5);
        a.y  = bandv(kk + 1 - m, w5);
        bb.x = Hh[kk][m];
        bb.y = Hh[kk + 1][m];
        acc = __builtin_amdgcn_wmma_f32_16x16x4_f32(
            /*neg_a=*/false, a, /*neg_b=*/false, bb,
            /*c_mod=*/(short)0, acc, /*reuse_a=*/false, /*reuse_b=*/false);
    }

    // D layout: lanes 0-15 -> rows vg, lanes 16-31 -> rows 8+vg; col = lane&15.
    float* dst = dens + (size_t)b * HW_;
#pragma unroll
    for (int vg = 0; vg < 8; ++vg) {
        const int row = r0 + vg + (hi << 3);
        const float s = acc[vg];
        dst[row * W_ + (c0 + m)] = 0.1f + 0.9f * sqrtf(fmaxf(s, 0.f));
    }
}

// ---------------- Kernel 3: streaming coords + weights (the 160 MB pass) -----
__global__ void __launch_bounds__(256)
k_emit(const float* __restrict__ dens, const float* __restrict__ rnd,
       float* __restrict__ coords, float* __restrict__ wts) {
    const size_t idx = (size_t)blockIdx.x * 256 + threadIdx.x; // exactly B*H*W threads
    const float local = dens[idx];
    const int pix = (int)(idx & (size_t)(HW_ - 1));
    const int i = pix >> 9, j = pix & 511;

    const int   num = local > 0.7f ? 4 : (local > 0.4f ? 2 : 1);
    const float yb  = -1.f + (2.f / 511.f) * (float)i;   // linspace(-1,1,512)
    const float xb  = -1.f + (2.f / 511.f) * (float)j;
    const float sc  = (num > 1) ? (2.f / 512.f) * 0.8f : 0.f; // jitter scale or 0

    const f4* rp = (const f4*)rnd + 2 * idx;             // [.. ,4,2] = 8 floats/pixel
    const f4 ra = __builtin_nontemporal_load(rp);
    const f4 rb = __builtin_nontemporal_load(rp + 1);

    f4 ca, cb;
    ca.x = yb + (ra.x - 0.5f) * sc;  ca.y = xb + (ra.y - 0.5f) * sc;
    ca.z = yb + (ra.z - 0.5f) * sc;  ca.w = xb + (ra.w - 0.5f) * sc;
    cb.x = yb + (rb.x - 0.5f) * sc;  cb.y = xb + (rb.y - 0.5f) * sc;
    cb.z = yb + (rb.z - 0.5f) * sc;  cb.w = xb + (rb.w - 0.5f) * sc;

    f4* cp = (f4*)coords + 2 * idx;
    __builtin_nontemporal_store(ca, cp);
    __builtin_nontemporal_store(cb, cp + 1);

    const float wv = local / (float)num;
    f4 w;
    w.x = wv;                        // slot 0 always valid (num>=1)
    w.y = (num > 1) ? wv : 0.f;
    w.z = (num > 2) ? wv : 0.f;      // num in {1,2,4}
    w.w = (num > 3) ? wv : 0.f;
    __builtin_nontemporal_store(w, (f4*)wts + idx);
}

// ---------------- launcher ---------------------------------------------------
extern "C" void kernel_launch(void* const* d_in, const int* in_sizes, int n_in,
                              void* d_out, int out_size, void* d_ws, size_t ws_size,
                              hipStream_t stream) {
    const float* contrast = (const float*)d_in[0];   // [8,1,512,512]
    const float* rnd      = (const float*)d_in[1];   // [8,512,512,4,2]
    const float* k2       = (const float*)d_in[2];   // [1,1,5,5]
    // d_in[3], d_in[4]: target_height/width == 512 (fixed by setup)

    float* coords  = (float*)d_out;                       // 16,777,216 f32
    float* weights = coords + (size_t)B_ * HW_ * 4;       // + 8,388,608 f32
    float* density = weights + (size_t)B_ * HW_ * 4 / 2;  // wrong? no:
    // coords = B*H*W*8, weights = B*H*W*4, density = B*H*W
    weights = coords + (size_t)B_ * HW_ * 8;
    density = weights + (size_t)B_ * HW_ * 4;

    float* ws = (float*)d_ws; // [0,512) pmin, [512,1024) pmax, [1024,1032) mn, [1032,1040) mx

    k_minmax_part<<<dim3(64, B_), 256, 0, stream>>>(contrast, ws, ws + 512);
    k_minmax_final<<<B_, 64, 0, stream>>>(ws);
    k_density<<<dim3(W_ / 16, H_ / 16, B_), 32, 0, stream>>>(contrast, k2, ws, density);
    k_emit<<<(B_ * HW_) / 256, 256, 0, stream>>>(density, rnd, coords, weights);
}